// NodeLevelSet_85280870629636
// MI455X (gfx1250) — compile-verified
//
#include <hip/hip_runtime.h>
#include <stdint.h>

#define MU_F 0.5f
#define CUTOFF 1e-10f

// ---------------------------------------------------------------------------
// Kernel 1: zero the grid-normal accumulator (d_ws), float4-vectorized stores.
// ---------------------------------------------------------------------------
__global__ void zero_normals(float* __restrict__ p, int n) {
  int i = (blockIdx.x * blockDim.x + threadIdx.x) * 4;
  if (i + 3 < n) {
    *(float4*)(p + i) = make_float4(0.f, 0.f, 0.f, 0.f);
  } else {
    for (int k = i; k < n; ++k) p[k] = 0.f;
  }
}

// ---------------------------------------------------------------------------
// Kernel 2: stream node_moment_nt -> d_out via the CDNA5 async LDS-DMA path.
// Each block owns a 16KB tile: every lane issues 4 outstanding
// global_load_async_to_lds_b128 (ASYNCcnt=4 in flight), one s_wait_asynccnt,
// then 4 global_store_async_from_lds_b128. No VGPR data movement.
// ---------------------------------------------------------------------------
__global__ void copy_moment_async(const float* __restrict__ src,
                                  float* __restrict__ dst, int n) {
  __shared__ __align__(16) float smem[256 * 16];   // 16 KB tile
  int tid = threadIdx.x;
  long long base = (long long)blockIdx.x * (256 * 16);

#pragma unroll
  for (int j = 0; j < 4; ++j) {
    long long i = base + (long long)(j * 256 + tid) * 4;
    if (i + 3 < n) {
      uint32_t lds = (uint32_t)(uintptr_t)(&smem[(j * 256 + tid) * 4]);
      uint64_t ga = (uint64_t)(uintptr_t)(src + i);
      asm volatile("global_load_async_to_lds_b128 %0, %1, off"
                   :: "v"(lds), "v"(ga) : "memory");
    }
  }
  asm volatile("s_wait_asynccnt 0" ::: "memory");
#pragma unroll
  for (int j = 0; j < 4; ++j) {
    long long i = base + (long long)(j * 256 + tid) * 4;
    if (i + 3 < n) {
      uint32_t lds = (uint32_t)(uintptr_t)(&smem[(j * 256 + tid) * 4]);
      uint64_t gb = (uint64_t)(uintptr_t)(dst + i);
      asm volatile("global_store_async_from_lds_b128 %0, %1, off"
                   :: "v"(gb), "v"(lds) : "memory");
    } else {
      for (long long k = i; k < n; ++k) dst[k] = src[k];  // ragged tail
    }
  }
  asm volatile("s_wait_asynccnt 0" ::: "memory");
}

// ---------------------------------------------------------------------------
// Kernel 3: P2G scatter. One thread per particle: 6x float4 grad loads,
// 2x int4 id loads, 24 no-return global_atomic_add_f32 (L2-resident target).
// ---------------------------------------------------------------------------
__global__ void p2g_scatter(const float* __restrict__ mass,
                            const float4* __restrict__ grads4,  // 6 per particle
                            const int4* __restrict__ ids4,      // 2 per particle
                            float* __restrict__ normals, int np) {
  int p = blockIdx.x * blockDim.x + threadIdx.x;
  if (p >= np) return;
  float m = mass[p];
  float4 g[6];
#pragma unroll
  for (int k = 0; k < 6; ++k) g[k] = grads4[(size_t)p * 6 + k];
  int4 idlo = ids4[(size_t)p * 2 + 0];
  int4 idhi = ids4[(size_t)p * 2 + 1];
  int ids[8] = {idlo.x, idlo.y, idlo.z, idlo.w, idhi.x, idhi.y, idhi.z, idhi.w};
  const float* gf = (const float*)g;
#pragma unroll
  for (int w = 0; w < 8; ++w) {
    float* base = normals + (size_t)ids[w] * 3;
    atomicAdd(base + 0, gf[w * 3 + 0] * m);
    atomicAdd(base + 1, gf[w * 3 + 1] * m);
    atomicAdd(base + 2, gf[w * 3 + 2] * m);
  }
}

// ---------------------------------------------------------------------------
// Kernel 4: per-selected-node friction update, overwrite rows of d_out.
// ---------------------------------------------------------------------------
__global__ void friction_update(const float* __restrict__ normals,  // ws [NC,3]
                                const float* __restrict__ moment,   // [NC,3]
                                const float* __restrict__ nmass,    // [NC]
                                const int* __restrict__ sel,        // [NS]
                                const float* __restrict__ vel,      // [NS,3]
                                float* __restrict__ out,            // [NC,3]
                                int ns) {
  int i = blockIdx.x * blockDim.x + threadIdx.x;
  if (i >= ns) return;
  int id = sel[i];
  size_t b = (size_t)id * 3;

  float nx = normals[b + 0], ny = normals[b + 1], nz = normals[b + 2];
  float mx = moment[b + 0],  my = moment[b + 1],  mz = moment[b + 2];
  float m = nmass[id];

  bool valid = m > CUTOFF;
  float inv = valid ? 1.0f / m : 0.0f;
  float vx = mx * inv, vy = my * inv, vz = mz * inv;  // vel_nt

  float nn = sqrtf(nx * nx + ny * ny + nz * nz);
  bool ok = valid && (nn > 0.f);
  float rn = ok ? 1.0f / nn : 0.0f;
  float hx = nx * rn, hy = ny * rn, hz = nz * rn;      // normal_hat

  float vsx = vel[(size_t)i * 3 + 0];
  float vsy = vel[(size_t)i * 3 + 1];
  float vsz = vel[(size_t)i * 3 + 2];
  float dx = vx - vsx, dy = vy - vsy, dz = vz - vsz;   // delta_vel
  float dvn = dx * hx + dy * hy + dz * hz;

  // cross = delta_vel x normal_hat
  float cx = dy * hz - dz * hy;
  float cy = dz * hx - dx * hz;
  float cz = dx * hy - dy * hx;
  float cn = sqrtf(cx * cx + cy * cy + cz * cz);
  float rc = (cn > 0.f) ? 1.0f / cn : 0.0f;
  float ox = cx * rc, oy = cy * rc, oz = cz * rc;      // omega

  float denom = (dvn != 0.f) ? dvn : 1.0f;
  float mu_p = fminf(MU_F, cn / denom);

  // tangent = hat + mu' * (hat x omega)
  float tx = hy * oz - hz * oy;
  float ty = hz * ox - hx * oz;
  float tz = hx * oy - hy * ox;
  tx = hx + mu_p * tx; ty = hy + mu_p * ty; tz = hz + mu_p * tz;

  float nvx = vx, nvy = vy, nvz = vz;
  if (dvn > 0.f) { nvx = vx - dvn * tx; nvy = vy - dvn * ty; nvz = vz - dvn * tz; }

  out[b + 0] = nvx * m;
  out[b + 1] = nvy * m;
  out[b + 2] = nvz * m;
}

// ---------------------------------------------------------------------------
extern "C" void kernel_launch(void* const* d_in, const int* in_sizes, int n_in,
                              void* d_out, int out_size, void* d_ws, size_t ws_size,
                              hipStream_t stream) {
  const float* mass   = (const float*)d_in[0];   // [NP]
  const float* grads  = (const float*)d_in[1];   // [NP,8,3]
  const int*   cids   = (const int*)d_in[2];     // [NP,8]
  const float* moment = (const float*)d_in[3];   // [NC,3]
  const float* nmass  = (const float*)d_in[4];   // [NC]
  const int*   sel    = (const int*)d_in[5];     // [NS]
  const float* vel    = (const float*)d_in[6];   // [NS,3]
  float* out = (float*)d_out;                    // [NC,3]
  float* normals = (float*)d_ws;                 // [NC,3] scratch accumulator

  int np = in_sizes[0];
  int nc = in_sizes[4];
  int ns = in_sizes[5];
  int n3 = nc * 3;

  const int T = 256;
  int n4 = (n3 + 3) / 4;
  int copy_blocks = (n3 + (T * 16) - 1) / (T * 16);  // 16 floats per thread

  zero_normals<<<(n4 + T - 1) / T, T, 0, stream>>>(normals, n3);
  copy_moment_async<<<copy_blocks, T, 0, stream>>>(moment, out, n3);
  p2g_scatter<<<(np + T - 1) / T, T, 0, stream>>>(
      mass, (const float4*)grads, (const int4*)cids, normals, np);
  friction_update<<<(ns + T - 1) / T, T, 0, stream>>>(
      normals, moment, nmass, sel, vel, out, ns);
}